// Encoder_13804024889559
// MI455X (gfx1250) — compile-verified
//
#include <hip/hip_runtime.h>
#include <hip/hip_bf16.h>

typedef __attribute__((ext_vector_type(2))) float v2f;
typedef __attribute__((ext_vector_type(4))) float v4f;
typedef __attribute__((ext_vector_type(8))) float v8f;

#define NPART 16384

// ---------------------------------------------------------------------------
// MLP kernel: one wave32 computes a 16-row tile with V_WMMA_F32_16X16X4_F32.
// A-fragment (16x4 f32): lane = M (mod 16), half-wave selects K pair.
// B/C/D: lane = N column, VGPR index + half-wave selects K / M row.
// ---------------------------------------------------------------------------
__global__ __launch_bounds__(256) void mlp_wmma_kernel(
    const float* __restrict__ pos,                       // (N, 6, 2)
    const float* __restrict__ W1, const float* __restrict__ b1,   // 12x32, 32
    const float* __restrict__ W2, const float* __restrict__ b2,   // 32x64, 64
    const float* __restrict__ W3, const float* __restrict__ b3,   // 64x128, 128
    float* __restrict__ vout)                            // (N, 128)
{
    __shared__ float xs[8][16 * 12];   // per-wave X staging
    __shared__ float hs[8][16 * 64];   // per-wave activation staging (max 16x64)

    const int wave  = threadIdx.x >> 5;
    const int lane  = threadIdx.x & 31;
    const int mrow  = lane & 15;       // M for A-frag, N column for B/D-frag
    const int khalf = lane >> 4;       // which K/M half this lane holds
    const int m0    = (blockIdx.x * 8 + wave) * 16;

    float* X = xs[wave];
    float* H = hs[wave];

    // ---- Stage X = [speeds(5x2), last(2)] (16 x 12) into LDS ----
    #pragma unroll
    for (int e = lane; e < 16 * 6; e += 32) {
        int m = e / 6, c2 = e % 6;
        const float* pr = pos + (size_t)(m0 + m) * 12;
        float vx, vy;
        if (c2 < 5) { vx = pr[(c2 + 1) * 2]     - pr[c2 * 2];
                      vy = pr[(c2 + 1) * 2 + 1] - pr[c2 * 2 + 1]; }
        else        { vx = pr[10]; vy = pr[11]; }
        X[m * 12 + c2 * 2]     = vx;
        X[m * 12 + c2 * 2 + 1] = vy;
    }
    __syncthreads();

    // ---- Layer 1: (16x12) @ (12x32) + b1, ReLU -> H (16x32) ----
    v8f a1[2];
    #pragma unroll
    for (int t = 0; t < 2; ++t) a1[t] = (v8f){0.f,0.f,0.f,0.f,0.f,0.f,0.f,0.f};
    #pragma unroll
    for (int k = 0; k < 3; ++k) {
        int kk = k * 4 + khalf * 2;
        v2f a; a.x = X[mrow * 12 + kk]; a.y = X[mrow * 12 + kk + 1];
        #pragma unroll
        for (int t = 0; t < 2; ++t) {
            v2f b; b.x = W1[kk * 32 + t * 16 + mrow];
                   b.y = W1[(kk + 1) * 32 + t * 16 + mrow];
            a1[t] = __builtin_amdgcn_wmma_f32_16x16x4_f32(
                false, a, false, b, (short)0, a1[t], false, false);
        }
    }
    #pragma unroll
    for (int t = 0; t < 2; ++t) {
        float bias = b1[t * 16 + mrow];
        #pragma unroll
        for (int r = 0; r < 8; ++r) {
            float h = a1[t][r] + bias;
            H[(r + khalf * 8) * 32 + t * 16 + mrow] = h > 0.f ? h : 0.f;
        }
    }
    __syncthreads();

    // ---- Layer 2: (16x32) @ (32x64) + b2, ReLU -> H (16x64) ----
    v8f a2[4];
    #pragma unroll
    for (int t = 0; t < 4; ++t) a2[t] = (v8f){0.f,0.f,0.f,0.f,0.f,0.f,0.f,0.f};
    #pragma unroll
    for (int k = 0; k < 8; ++k) {
        int kk = k * 4 + khalf * 2;
        v2f a; a.x = H[mrow * 32 + kk]; a.y = H[mrow * 32 + kk + 1];
        #pragma unroll
        for (int t = 0; t < 4; ++t) {
            v2f b; b.x = W2[kk * 64 + t * 16 + mrow];
                   b.y = W2[(kk + 1) * 64 + t * 16 + mrow];
            a2[t] = __builtin_amdgcn_wmma_f32_16x16x4_f32(
                false, a, false, b, (short)0, a2[t], false, false);
        }
    }
    __syncthreads();   // all reads of H(16x32) done before overwrite as 16x64
    #pragma unroll
    for (int t = 0; t < 4; ++t) {
        float bias = b2[t * 16 + mrow];
        #pragma unroll
        for (int r = 0; r < 8; ++r) {
            float h = a2[t][r] + bias;
            H[(r + khalf * 8) * 64 + t * 16 + mrow] = h > 0.f ? h : 0.f;
        }
    }
    __syncthreads();

    // ---- Layer 3: (16x64) @ (64x128) + b3 -> vout ----
    v8f a3[8];
    #pragma unroll
    for (int t = 0; t < 8; ++t) a3[t] = (v8f){0.f,0.f,0.f,0.f,0.f,0.f,0.f,0.f};
    #pragma unroll
    for (int k = 0; k < 16; ++k) {
        int kk = k * 4 + khalf * 2;
        v2f a; a.x = H[mrow * 64 + kk]; a.y = H[mrow * 64 + kk + 1];
        #pragma unroll
        for (int t = 0; t < 8; ++t) {
            v2f b; b.x = W3[kk * 128 + t * 16 + mrow];
                   b.y = W3[(kk + 1) * 128 + t * 16 + mrow];
            a3[t] = __builtin_amdgcn_wmma_f32_16x16x4_f32(
                false, a, false, b, (short)0, a3[t], false, false);
        }
    }
    #pragma unroll
    for (int t = 0; t < 8; ++t) {
        float bias = b3[t * 16 + mrow];
        #pragma unroll
        for (int r = 0; r < 8; ++r)
            vout[(size_t)(m0 + r + khalf * 8) * 128 + t * 16 + mrow] = a3[t][r] + bias;
    }
}

// ---------------------------------------------------------------------------
// Adjacency: adj[i][j] = (sq_i + sq_j - 2*dot <= R^2) && i != j, as 0/1 float.
// 1.07 GB of stores -> pure HBM-store-bound. Each thread owns 4 consecutive
// columns and emits one nontemporal global_store_b128 per row: 512 B per
// wave-store-instruction, perfectly coalesced, no L2 pollution.
// Block = 32 rows x 4096-col strip; p_i tile in LDS, p_j reused across rows.
// ---------------------------------------------------------------------------
__global__ __launch_bounds__(256) void adj_kernel(
    const float* __restrict__ pos, float* __restrict__ adj)
{
    __shared__ float pix[32], piy[32], pisq[32];
    const int i0 = blockIdx.x * 32;
    if (threadIdx.x < 32) {
        int i = i0 + threadIdx.x;
        float x = pos[(size_t)i * 12 + 10];
        float y = pos[(size_t)i * 12 + 11];
        pix[threadIdx.x] = x; piy[threadIdx.x] = y; pisq[threadIdx.x] = x * x + y * y;
    }
    __syncthreads();

    const int j0 = blockIdx.y * 4096;
    for (int jb = j0 + (int)threadIdx.x * 4; jb < j0 + 4096; jb += 1024) {
        float xj[4], yj[4], sqj[4];
        #pragma unroll
        for (int q = 0; q < 4; ++q) {
            xj[q]  = pos[(size_t)(jb + q) * 12 + 10];
            yj[q]  = pos[(size_t)(jb + q) * 12 + 11];
            sqj[q] = xj[q] * xj[q] + yj[q] * yj[q];
        }
        #pragma unroll
        for (int r = 0; r < 32; ++r) {
            int i = i0 + r;
            v4f out;
            #pragma unroll
            for (int q = 0; q < 4; ++q) {
                float d2 = pisq[r] + sqj[q]
                         - 2.0f * (pix[r] * xj[q] + piy[r] * yj[q]);
                out[q] = (d2 <= 4.0f && i != (jb + q)) ? 1.0f : 0.0f;
            }
            __builtin_nontemporal_store(
                out, (v4f*)(adj + (size_t)i * NPART + jb));
        }
    }
}

extern "C" void kernel_launch(void* const* d_in, const int* in_sizes, int n_in,
                              void* d_out, int out_size, void* d_ws, size_t ws_size,
                              hipStream_t stream) {
    (void)in_sizes; (void)n_in; (void)out_size; (void)d_ws; (void)ws_size;
    const float* pos = (const float*)d_in[0];
    const float* W1  = (const float*)d_in[1];
    const float* b1  = (const float*)d_in[2];
    const float* W2  = (const float*)d_in[3];
    const float* b2  = (const float*)d_in[4];
    const float* W3  = (const float*)d_in[5];
    const float* b3  = (const float*)d_in[6];
    // d_in[7] = e0 is unused by the reference.

    float* v   = (float*)d_out;                 // N x 128
    float* adj = v + (size_t)NPART * 128;       // N x N (0/1 floats)

    mlp_wmma_kernel<<<dim3(NPART / 128), dim3(256), 0, stream>>>(
        pos, W1, b1, W2, b2, W3, b3, v);
    adj_kernel<<<dim3(NPART / 32, 4), dim3(256), 0, stream>>>(pos, adj);
}